// Encoder_67705864454764
// MI455X (gfx1250) — compile-verified
//
#include <hip/hip_runtime.h>
#include <hip/hip_bf16.h>

// ---------------- problem constants ----------------
#define BATCH     8
#define NS        32768
#define KERN      256
#define DSZ       256            // number of atoms
#define TPOS      (NS - KERN + 1) // 32513 valid positions
#define NSTEP     16
#define CH        128

// ---------------- tiling ----------------
#define NT        64             // positions per workgroup
#define NBLK      ((TPOS + NT - 1) / NT)   // 509
#define KC        64             // K-chunk staged in LDS
#define DPITCH    68             // LDS pitch for d chunk (float4-aligned, conflict-free)

typedef float v2f __attribute__((ext_vector_type(2)));
typedef float v8f __attribute__((ext_vector_type(8)));

// ---------------------------------------------------------------------------
// residual init: res = x
// ---------------------------------------------------------------------------
__global__ void mp_init(const float* __restrict__ x, float* __restrict__ res, int n) {
    int i = blockIdx.x * blockDim.x + threadIdx.x;
    if (i < n) res[i] = x[i];
}

// ---------------------------------------------------------------------------
// fused correlation (f32 WMMA) + per-workgroup argmax of |fm|
// grid: (NBLK, BATCH), block: 256 threads (8 waves of 32)
// Each WG: batch b, positions [n0, n0+NT), all 256 atoms.
// GEMM: C[a,n] = sum_k d[a,k] * r[n0+n+k],  M=256, N=NT, K=256
// wave w owns M-tiles {2w, 2w+1} x N-tiles {0..3}; 16x16x4 f32 WMMA.
// ---------------------------------------------------------------------------
__global__ void __launch_bounds__(256)
mp_corr_argmax(const float* __restrict__ res, const float* __restrict__ dG,
               float* __restrict__ pabs, float* __restrict__ pval,
               unsigned* __restrict__ pidx) {
    __shared__ float rbuf[NT + KERN];          // residual window: r[n0 .. n0+319]
    __shared__ float dbuf[DSZ * DPITCH];       // d chunk: 256 rows x KC cols (pitch 68)
    __shared__ float swabs[8], swval[8];
    __shared__ unsigned swidx[8];

    const int tid  = threadIdx.x;
    const int lane = tid & 31;
    const int wid  = tid >> 5;
    const int nloc = lane & 15;   // column within 16-wide tile / row within A tile
    const int kh   = lane >> 4;   // K-half selector for A/B fragments
    const int b    = blockIdx.y;
    const int n0   = blockIdx.x * NT;

    // load residual window (zero-pad past end of signal)
    for (int i = tid; i < NT + KERN; i += 256) {
        int g = n0 + i;
        rbuf[i] = (g < NS) ? res[b * NS + g] : 0.0f;
    }

    v8f acc[2][4];
#pragma unroll
    for (int mi = 0; mi < 2; ++mi)
#pragma unroll
        for (int ni = 0; ni < 4; ++ni)
            acc[mi][ni] = (v8f){0.f, 0.f, 0.f, 0.f, 0.f, 0.f, 0.f, 0.f};

    for (int kc0 = 0; kc0 < KERN; kc0 += KC) {
        __syncthreads();
        // stage d[:, kc0:kc0+KC] into LDS, float4 coalesced
#pragma unroll
        for (int i = 0; i < 16; ++i) {
            int e4 = i * 256 + tid;        // 4096 float4 elements
            int a  = e4 >> 4;              // row (atom)
            int c4 = e4 & 15;              // float4 column within chunk
            float4 v = *(const float4*)(dG + a * KERN + kc0 + c4 * 4);
            *(float4*)(&dbuf[a * DPITCH + c4 * 4]) = v;
        }
        __syncthreads();

        for (int k4 = 0; k4 < KC; k4 += 4) {
            // B fragments: B[k][n] = r[n0+n+k]  ->  rbuf[n + k]
            v2f bf[4];
            const int kg = kc0 + k4 + 2 * kh;  // global k for this lane's half
#pragma unroll
            for (int ni = 0; ni < 4; ++ni) {
                int nn = ni * 16 + nloc;
                bf[ni].x = rbuf[nn + kg];
                bf[ni].y = rbuf[nn + kg + 1];
            }
#pragma unroll
            for (int mi = 0; mi < 2; ++mi) {
                int m  = wid * 32 + mi * 16 + nloc;
                int kl = k4 + 2 * kh;
                v2f af;
                af.x = dbuf[m * DPITCH + kl];
                af.y = dbuf[m * DPITCH + kl + 1];
#pragma unroll
                for (int ni = 0; ni < 4; ++ni)
                    acc[mi][ni] = __builtin_amdgcn_wmma_f32_16x16x4_f32(
                        false, af, false, bf[ni], (short)0, acc[mi][ni],
                        false, false);
            }
        }
    }

    // ---- per-thread argmax over owned C elements ----
    float    best_abs = -1.0f, best_val = 0.0f;
    unsigned best_idx = 0xFFFFFFFFu;
#pragma unroll
    for (int mi = 0; mi < 2; ++mi)
#pragma unroll
        for (int ni = 0; ni < 4; ++ni)
#pragma unroll
            for (int r = 0; r < 8; ++r) {
                float v = acc[mi][ni][r];
                int atom = wid * 32 + mi * 16 + r + 8 * kh;
                int pos  = n0 + ni * 16 + nloc;
                if (pos < TPOS) {
                    float    a   = fabsf(v);
                    unsigned idx = (unsigned)atom * (unsigned)TPOS + (unsigned)pos;
                    if (a > best_abs || (a == best_abs && idx < best_idx)) {
                        best_abs = a; best_val = v; best_idx = idx;
                    }
                }
            }

    // ---- wave32 shuffle reduction ----
#pragma unroll
    for (int off = 16; off > 0; off >>= 1) {
        float    oa = __shfl_xor(best_abs, off, 32);
        float    ov = __shfl_xor(best_val, off, 32);
        unsigned oi = __shfl_xor(best_idx, off, 32);
        if (oa > best_abs || (oa == best_abs && oi < best_idx)) {
            best_abs = oa; best_val = ov; best_idx = oi;
        }
    }
    if (lane == 0) { swabs[wid] = best_abs; swval[wid] = best_val; swidx[wid] = best_idx; }
    __syncthreads();
    if (tid == 0) {
        float    ba = swabs[0], bv = swval[0];
        unsigned bi = swidx[0];
#pragma unroll
        for (int w = 1; w < 8; ++w) {
            if (swabs[w] > ba || (swabs[w] == ba && swidx[w] < bi)) {
                ba = swabs[w]; bv = swval[w]; bi = swidx[w];
            }
        }
        int p = b * NBLK + blockIdx.x;
        pabs[p] = ba; pval[p] = bv; pidx[p] = bi;
    }
}

// ---------------------------------------------------------------------------
// per-batch: reduce partials -> global argmax; subtract amp*d[atom] at pos;
// record keypoint. grid: BATCH blocks x 256 threads.
// ---------------------------------------------------------------------------
__global__ void __launch_bounds__(256)
mp_select_update(float* __restrict__ res, const float* __restrict__ dG,
                 const float* __restrict__ pabs, const float* __restrict__ pval,
                 const unsigned* __restrict__ pidx,
                 int* __restrict__ kppos, int* __restrict__ kpatom,
                 float* __restrict__ kpamp, int step) {
    __shared__ float    sabs[256], sval[256];
    __shared__ unsigned sidx[256];

    const int tid = threadIdx.x;
    const int b   = blockIdx.x;

    float    ba = -1.0f, bv = 0.0f;
    unsigned bi = 0xFFFFFFFFu;
    for (int i = tid; i < NBLK; i += 256) {
        float    a = pabs[b * NBLK + i];
        unsigned x = pidx[b * NBLK + i];
        if (a > ba || (a == ba && x < bi)) { ba = a; bv = pval[b * NBLK + i]; bi = x; }
    }
    sabs[tid] = ba; sval[tid] = bv; sidx[tid] = bi;
    __syncthreads();
    for (int s = 128; s > 0; s >>= 1) {
        if (tid < s) {
            if (sabs[tid + s] > sabs[tid] ||
                (sabs[tid + s] == sabs[tid] && sidx[tid + s] < sidx[tid])) {
                sabs[tid] = sabs[tid + s];
                sval[tid] = sval[tid + s];
                sidx[tid] = sidx[tid + s];
            }
        }
        __syncthreads();
    }

    unsigned idx  = sidx[0];
    float    amp  = sval[0];
    int      atom = (int)(idx / (unsigned)TPOS);
    int      pos  = (int)(idx - (unsigned)atom * (unsigned)TPOS);

    if (tid == 0) {
        kppos[b * NSTEP + step]  = pos;
        kpatom[b * NSTEP + step] = atom;
        kpamp[b * NSTEP + step]  = amp;
    }
    // subtract amp * d[atom] from residual at pos (KERN == blockDim)
    res[b * NS + pos + tid] -= amp * dG[atom * KERN + tid];
}

// ---------------------------------------------------------------------------
// final 1x1 convs; atoms channel is one-hot so at[c] = w_atoms[c,atom]+b.
// out[b][o][t], 8*128*16 threads.
// ---------------------------------------------------------------------------
__global__ void __launch_bounds__(256)
mp_head(const int* __restrict__ kppos, const int* __restrict__ kpatom,
        const float* __restrict__ kpamp,
        const float* __restrict__ w_pa, const float* __restrict__ b_pa,
        const float* __restrict__ w_at, const float* __restrict__ b_at,
        const float* __restrict__ w_rd, const float* __restrict__ b_rd,
        float* __restrict__ out) {
    int tid = blockIdx.x * blockDim.x + threadIdx.x;
    if (tid >= BATCH * CH * NSTEP) return;
    int t = tid & (NSTEP - 1);
    int o = (tid >> 4) & (CH - 1);
    int b = tid >> 11;

    float posf = (float)kppos[b * NSTEP + t] / (float)TPOS;
    float amp  = kpamp[b * NSTEP + t];
    int   atom = kpatom[b * NSTEP + t];

    float acc = b_rd[o];
    for (int c = 0; c < CH; ++c) {
        float h1 = w_pa[c * 2 + 0] * posf + w_pa[c * 2 + 1] * amp + b_pa[c];
        float h2 = w_at[c * DSZ + atom] + b_at[c];
        acc += w_rd[o * (2 * CH) + c] * h1 + w_rd[o * (2 * CH) + CH + c] * h2;
    }
    out[tid] = acc;
}

// ---------------------------------------------------------------------------
extern "C" void kernel_launch(void* const* d_in, const int* in_sizes, int n_in,
                              void* d_out, int out_size, void* d_ws, size_t ws_size,
                              hipStream_t stream) {
    const float* x     = (const float*)d_in[0];
    const float* dG    = (const float*)d_in[1];
    const float* w_pa  = (const float*)d_in[2];
    const float* b_pa  = (const float*)d_in[3];
    const float* w_at  = (const float*)d_in[4];
    const float* b_at  = (const float*)d_in[5];
    const float* w_rd  = (const float*)d_in[6];
    const float* b_rd  = (const float*)d_in[7];
    float*       out   = (float*)d_out;

    // workspace layout (floats)
    float*    res    = (float*)d_ws;                        // BATCH*NS
    float*    pabs   = res + BATCH * NS;                    // BATCH*NBLK
    float*    pval   = pabs + BATCH * NBLK;                 // BATCH*NBLK
    unsigned* pidx   = (unsigned*)(pval + BATCH * NBLK);    // BATCH*NBLK
    int*      kppos  = (int*)(pidx + BATCH * NBLK);         // BATCH*NSTEP
    int*      kpatom = kppos + BATCH * NSTEP;               // BATCH*NSTEP
    float*    kpamp  = (float*)(kpatom + BATCH * NSTEP);    // BATCH*NSTEP

    mp_init<<<(BATCH * NS + 255) / 256, 256, 0, stream>>>(x, res, BATCH * NS);

    for (int s = 0; s < NSTEP; ++s) {
        mp_corr_argmax<<<dim3(NBLK, BATCH), 256, 0, stream>>>(res, dG, pabs, pval, pidx);
        mp_select_update<<<BATCH, 256, 0, stream>>>(res, dG, pabs, pval, pidx,
                                                    kppos, kpatom, kpamp, s);
    }

    mp_head<<<(BATCH * CH * NSTEP + 255) / 256, 256, 0, stream>>>(
        kppos, kpatom, kpamp, w_pa, b_pa, w_at, b_at, w_rd, b_rd, out);
}